// LearnedClassVectors_51857435132562
// MI455X (gfx1250) — compile-verified
//
#include <hip/hip_runtime.h>
#include <hip/hip_bf16.h>

// CDNA5 / gfx1250: bucketize + embedding gather + patch-permute.
// Input x read as coalesced b128 (one float4 = 4 pw voxels per thread),
// output scattered by the Tensor Data Mover (tensor_store_from_lds).

typedef unsigned int u32x4 __attribute__((ext_vector_type(4)));
typedef int          i32x8 __attribute__((ext_vector_type(8)));
typedef int          i32x4 __attribute__((ext_vector_type(4)));

#define NVOX     128          // x is (B,1,128,128,128)
#define G        32           // 128/4 patches per axis
#define VDIM     12
#define CH_TOTAL 768          // 4*4*4*12
#define CH_HALF  384          // channels per TDM store (49,152B LDS tile)
#define C_STRIDE 32768        // out stride between channels = 32*32*32

__device__ __forceinline__ int hu_bin(float xv) {
    // searchsorted(edges, x, side='right') == count(edges <= x)
    int bin = 0;
    bin += (xv >= -1000.0f);
    bin += (xv >=   -75.0f);
    bin += (xv >=     0.0f);
    bin += (xv >=    15.0f);
    bin += (xv >=    25.0f);
    bin += (xv >=    40.0f);
    bin += (xv >=    50.0f);
    bin += (xv >=   200.0f);
    bin += (xv >=  1000.0f);
    return bin;
}

__global__ __launch_bounds__(256)
void lcv_tdm_kernel(const float* __restrict__ x,
                    const float* __restrict__ vectors,
                    float* __restrict__ out)
{
    // vectors rows padded 12 -> 16 floats so float4 gathers are 16B aligned
    __shared__ __align__(16) float vlds[10 * 16];
    // one half-tile: 384 channels x 32 gw, channel-major (TDM reads LDS linearly)
    __shared__ __align__(16) float tile[CH_HALF * G];

    const int t   = threadIdx.x;
    const unsigned blk = blockIdx.x;       // 2048 blocks = b(2) * gd(32) * gh(32)
    const int b  = blk >> 10;
    const int gd = (blk >> 5) & 31;
    const int gh = blk & 31;

    if (t < 120) {
        int r = t / VDIM;
        int c = t - r * VDIM;
        vlds[r * 16 + c] = vectors[t];
    }
    __syncthreads();

    const float* xb = x + (size_t)b * (NVOX * NVOX * NVOX);

    const int gw  = t & 31;       // lane-consecutive -> coalesced b128 loads
    const int pdh = t >> 5;       // 0..7 : (pd_local, ph)
    const int pdl = pdh >> 2;     // 0..1
    const int ph  = pdh & 3;

    for (int half = 0; half < 2; ++half) {
        const int d = gd * 4 + half * 2 + pdl;
        const int h = gh * 4 + ph;

        // one aligned 16B load = x for pw = 0..3 at this (d,h,gw)
        const float4 xq =
            *(const float4*)&xb[((size_t)d * NVOX + h) * NVOX + (size_t)gw * 4];

        #pragma unroll
        for (int pw = 0; pw < 4; ++pw) {
            const float xv = (pw == 0) ? xq.x : (pw == 1) ? xq.y
                           : (pw == 2) ? xq.z : xq.w;
            const int bin = hu_bin(xv);

            const float4 r0 = *(const float4*)&vlds[bin * 16 + 0];
            const float4 r1 = *(const float4*)&vlds[bin * 16 + 4];
            const float4 r2 = *(const float4*)&vlds[bin * 16 + 8];

            // local patchpos within this half: ppl = pdh*4 + pw
            // tile[(ppl*12 + v)*32 + gw]; lane-consecutive gw -> conflict-free
            const int base = (pdh * 4 + pw) * (VDIM * G) + gw;
            tile[base +  0 * G] = r0.x;
            tile[base +  1 * G] = r0.y;
            tile[base +  2 * G] = r0.z;
            tile[base +  3 * G] = r0.w;
            tile[base +  4 * G] = r1.x;
            tile[base +  5 * G] = r1.y;
            tile[base +  6 * G] = r1.z;
            tile[base +  7 * G] = r1.w;
            tile[base +  8 * G] = r2.x;
            tile[base +  9 * G] = r2.y;
            tile[base + 10 * G] = r2.z;
            tile[base + 11 * G] = r2.w;
        }
        __syncthreads();   // all ds_stores visible before TDM reads LDS

        if (t < 32) {      // wave 0 issues one TDM store (EXEC ignored, per-wave op)
            // out element offset of (b, c0=half*384, gd, gh, gw=0)
            const size_t elem = (size_t)(b * CH_TOTAL + half * CH_HALF) * C_STRIDE
                              + (size_t)gd * (G * G) + (size_t)gh * G;
            const unsigned long long gaddr =
                (unsigned long long)(uintptr_t)(out + elem);
            const unsigned lds_off = (unsigned)(uintptr_t)&tile[0]; // low 32b = LDS offset

            // D# group 0: count=1 | lds_addr | global_addr[56:0] | type=2
            u32x4 g0;
            g0[0] = 1u;
            g0[1] = lds_off;
            g0[2] = (unsigned)gaddr;
            g0[3] = ((unsigned)(gaddr >> 32) & 0x01FFFFFFu) | 0x80000000u;

            // D# group 1: 2D tile, data_size=4B
            //  rows of 32 contiguous f32 (128B-aligned full lines),
            //  384 rows, row stride 32768 elems
            i32x8 g1;
            g1[0] = 0x00020000;                     // workgroup_mask=0, data_size=2(4B)
            g1[1] = (int)(32u << 16);               // tensor_dim0[15:0]=32
            g1[2] = (int)((unsigned)CH_HALF << 16); // tensor_dim1[15:0]=384
            g1[3] = (int)(32u << 16);               // tile_dim0=32
            g1[4] = CH_HALF;                        // tile_dim1=384, tile_dim2=0
            g1[5] = C_STRIDE;                       // tensor_dim0_stride[31:0]=32768
            g1[6] = 0;                              // stride hi / dim1_stride lo
            g1[7] = 0;

            i32x4 z4 = {0, 0, 0, 0};                 // groups 2/3 unused (2D tensor)
            i32x8 z8 = {0, 0, 0, 0, 0, 0, 0, 0};     // extra group (clang-23 6-arg form)
            __builtin_amdgcn_tensor_store_from_lds(g0, g1, z4, z4, z8, 0);
            __builtin_amdgcn_s_wait_tensorcnt(0);    // tile reused next iteration
        }
        __syncthreads();
    }
}

extern "C" void kernel_launch(void* const* d_in, const int* in_sizes, int n_in,
                              void* d_out, int out_size, void* d_ws, size_t ws_size,
                              hipStream_t stream) {
    (void)in_sizes; (void)n_in; (void)d_ws; (void)ws_size; (void)out_size;
    const float* x    = (const float*)d_in[0];   // (2,1,128,128,128) f32
    const float* vecs = (const float*)d_in[1];   // (10,12) f32
    float* out        = (float*)d_out;           // (2,768,32,32,32) f32

    dim3 grid(2 * G * G);   // 2048 blocks: (b, gd, gh)
    dim3 block(256);
    hipLaunchKernelGGL(lcv_tdm_kernel, grid, block, 0, stream, x, vecs, out);
}